// ScaleInvariantReconstructionLoss_18734647345452
// MI455X (gfx1250) — compile-verified
//
#include <hip/hip_runtime.h>

// ---------------------------------------------------------------------------
// ScaleInvariantReconstructionLoss for MI455X (gfx1250, wave32)
//
// term1 = 1 - (sum over all b,c,h,w of q*t) / (B*H*W)          [131072]
// term2 = sum_{b,n} max( dot_w( T[b,idx1[n],:], T[7-b,idx2[n],:] ), 0 ) / 512
// out   = term1 + term2   (scalar fp32)
//
// Bandwidth-bound: 256 MiB single-pass @ 23.3 TB/s ~ 11 us floor.
// Uses non-temporal b128 loads + V_WMMA_F32_16X16X4_F32 as the wave reducer.
// ---------------------------------------------------------------------------

typedef __attribute__((ext_vector_type(2))) float v2f;
typedef __attribute__((ext_vector_type(4))) float v4f;
typedef __attribute__((ext_vector_type(8))) float v8f;

#define B_DIM 8
#define C_DIM 256
#define H_DIM 128
#define W_DIM 128
#define CH (C_DIM * H_DIM)                         // 32768 rows per batch
#define TOTAL_ELEMS (B_DIM * C_DIM * H_DIM * W_DIM) // 33554432
#define TOTAL_V4 (TOTAL_ELEMS / 4)                 // 8388608 float4 pairs
#define DOT_BLOCKS 2048
#define DOT_THREADS 256
#define DOT_TOTAL (DOT_BLOCKS * DOT_THREADS)       // 524288 threads
#define DOT_ITERS (TOTAL_V4 / DOT_TOTAL)           // 16, exact (no tail)

// --- Kernel 1: global dot(query,target), memory-bound streaming reduction ---
__global__ __launch_bounds__(DOT_THREADS) void sirl_dot_kernel(
    const v4f* __restrict__ q, const v4f* __restrict__ t, float* __restrict__ acc) {
  unsigned tid = blockIdx.x * DOT_THREADS + threadIdx.x;

  // Two fp32 accumulators per lane -> exactly the 2 A-matrix VGPRs (16x4 f32).
  float a0 = 0.0f, a1 = 0.0f;
  size_t idx = tid;
#pragma unroll 4
  for (int i = 0; i < DOT_ITERS; ++i) {
    v4f qv = __builtin_nontemporal_load(&q[idx]);   // TH=NT: no reuse, spare L2
    v4f tv = __builtin_nontemporal_load(&t[idx]);
    a0 = fmaf(qv.x, tv.x, a0);
    a0 = fmaf(qv.y, tv.y, a0);
    a1 = fmaf(qv.z, tv.z, a1);
    a1 = fmaf(qv.w, tv.w, a1);
    idx += DOT_TOTAL;
  }

  // Wave-level reduction via WMMA: A(16x4) holds all 64 per-wave partials,
  // B = all-ones (layout independent), C = 0.
  // D[m][n] = rowsum(A[m]) replicated across n, so per-lane sum(d0..d7) is the
  // half-wave total; xor-16 shuffle completes the full wave sum exactly.
  // EXEC is all-ones here (no divergence, exact trip counts) as WMMA requires.
  v2f A;  A.x = a0;   A.y = a1;
  v2f Bm; Bm.x = 1.0f; Bm.y = 1.0f;
  v8f C = {};
  v8f D = __builtin_amdgcn_wmma_f32_16x16x4_f32(
      /*neg_a=*/false, A, /*neg_b=*/false, Bm,
      /*c_mod=*/(short)0, C, /*reuse_a=*/false, /*reuse_b=*/false);
  float s = ((D[0] + D[1]) + (D[2] + D[3])) + ((D[4] + D[5]) + (D[6] + D[7]));
  s += __shfl_xor(s, 16, 32);                       // combine the two M-halves

  if ((threadIdx.x & 31u) == 0u)
    atomicAdd(acc, s);                              // global_atomic_add_f32
}

// --- Kernel 2: 512 row-dot products (one wave32 per (b,n) pair) -------------
__global__ __launch_bounds__(256) void sirl_sim_kernel(
    const float* __restrict__ t, const int* __restrict__ idx1,
    const int* __restrict__ idx2, float* __restrict__ acc) {
  unsigned gtid = blockIdx.x * 256u + threadIdx.x;
  unsigned wave = gtid >> 5;     // 0..511
  unsigned lane = gtid & 31u;
  unsigned b = wave >> 6;        // 0..7
  unsigned n = wave & 63u;

  int r1 = idx1[n];
  int r2 = idx2[n];
  const v4f* p1 = (const v4f*)(t + ((size_t)b * CH + (size_t)r1) * W_DIM);
  const v4f* p2 = (const v4f*)(t + ((size_t)(B_DIM - 1 - b) * CH + (size_t)r2) * W_DIM);

  v4f x = p1[lane];              // lane covers w = 4*lane .. 4*lane+3 (W=128)
  v4f y = p2[lane];
  float s = fmaf(x.x, y.x, fmaf(x.y, y.y, fmaf(x.z, y.z, x.w * y.w)));
  s += __shfl_xor(s, 16, 32);
  s += __shfl_xor(s, 8, 32);
  s += __shfl_xor(s, 4, 32);
  s += __shfl_xor(s, 2, 32);
  s += __shfl_xor(s, 1, 32);

  if (lane == 0u)
    atomicAdd(acc, fmaxf(s, 0.0f));
}

// --- Kernel 3: combine the two scalar accumulators --------------------------
__global__ void sirl_finalize_kernel(const float* __restrict__ ws,
                                     float* __restrict__ out) {
  out[0] = 1.0f - ws[0] * (1.0f / 131072.0f) + ws[1] * (1.0f / 512.0f);
}

extern "C" void kernel_launch(void* const* d_in, const int* in_sizes, int n_in,
                              void* d_out, int out_size, void* d_ws, size_t ws_size,
                              hipStream_t stream) {
  const v4f*   q    = (const v4f*)d_in[0];
  const v4f*   t4   = (const v4f*)d_in[1];
  const float* tf   = (const float*)d_in[1];
  const int*   idx1 = (const int*)d_in[2];
  const int*   idx2 = (const int*)d_in[3];
  float*       out  = (float*)d_out;
  float*       ws   = (float*)d_ws;

  // ws[0] = dot accumulator, ws[1] = relu(similarity) accumulator.
  // Must re-zero every call (harness does not re-poison between replays).
  hipMemsetAsync(ws, 0, 2 * sizeof(float), stream);

  sirl_dot_kernel<<<DOT_BLOCKS, DOT_THREADS, 0, stream>>>(q, t4, ws);
  sirl_sim_kernel<<<64, 256, 0, stream>>>(tf, idx1, idx2, ws + 1);
  sirl_finalize_kernel<<<1, 1, 0, stream>>>(ws, out);
}